// AgentUpdate_59193239274121
// MI455X (gfx1250) — compile-verified
//
#include <hip/hip_runtime.h>
#include <hip/hip_bf16.h>
#include <cstdint>

// ---------------- Threefry-2x32 (JAX-compatible), constexpr-capable ----------------
struct U2 { unsigned a, b; };

__host__ __device__ constexpr unsigned rotl32(unsigned v, int n) {
    return (v << n) | (v >> (32 - n));
}

__host__ __device__ constexpr U2 tf2x32(unsigned k0, unsigned k1, unsigned x0, unsigned x1) {
    const unsigned k2 = k0 ^ k1 ^ 0x1BD11BDAu;
#define TF_ROUND(R) x0 += x1; x1 = rotl32(x1, R); x1 ^= x0;
    x0 += k0; x1 += k1;
    TF_ROUND(13) TF_ROUND(15) TF_ROUND(26) TF_ROUND(6)
    x0 += k1; x1 += k2 + 1u;
    TF_ROUND(17) TF_ROUND(29) TF_ROUND(16) TF_ROUND(24)
    x0 += k2; x1 += k0 + 2u;
    TF_ROUND(13) TF_ROUND(15) TF_ROUND(26) TF_ROUND(6)
    x0 += k0; x1 += k1 + 3u;
    TF_ROUND(17) TF_ROUND(29) TF_ROUND(16) TF_ROUND(24)
    x0 += k1; x1 += k2 + 4u;
    TF_ROUND(13) TF_ROUND(15) TF_ROUND(26) TF_ROUND(6)
    x0 += k2; x1 += k0 + 5u;
#undef TF_ROUND
    return U2{x0, x1};
}

// key(42) = [0,42]; fold_in(key, d) = threefry(key, [0, d])  (computed at compile time)
constexpr U2 KCAT = tf2x32(0u, 42u, 0u, 1u);  // categorical sampling key
constexpr U2 KTH  = tf2x32(0u, 42u, 0u, 2u);  // theta-resample key

__device__ __forceinline__ float u01(unsigned bits) {
    // JAX: bitcast((bits >> 9) | 0x3f800000) - 1.0  ->  [0, 1)
    return __uint_as_float((bits >> 9) | 0x3f800000u) - 1.0f;
}

#define BLK 256

// Address-space-typed pointers for the CDNA5 async VMEM->LDS builtins.
typedef __attribute__((address_space(1))) int as1_int;
typedef __attribute__((address_space(3))) int as3_int;

__global__ __launch_bounds__(BLK)
void AgentUpdate_kernel(const float* __restrict__ gx,
                        const float* __restrict__ gy,
                        const float* __restrict__ gth,
                        const float* __restrict__ frame,
                        float* __restrict__ ox,
                        float* __restrict__ oy,
                        float* __restrict__ oth,
                        int n) {
    const unsigned i = blockIdx.x * BLK + threadIdx.x;
    if (i >= (unsigned)n) return;

#if __has_builtin(__builtin_amdgcn_global_load_async_to_lds_b32) && __has_builtin(__builtin_amdgcn_s_wait_asynccnt)
    #define USE_ASYNC_LDS 1
#else
    #define USE_ASYNC_LDS 0
#endif

#if USE_ASYNC_LDS
    // -------- Stage 1: launch async VMEM->LDS streaming loads (ASYNCcnt) --------
    __shared__ float tx[BLK], ty[BLK], tt[BLK];
    __builtin_amdgcn_global_load_async_to_lds_b32(
        (as1_int*)(gx + i), (as3_int*)&tx[threadIdx.x], 0, 0);
    __builtin_amdgcn_global_load_async_to_lds_b32(
        (as1_int*)(gy + i), (as3_int*)&ty[threadIdx.x], 0, 0);
    __builtin_amdgcn_global_load_async_to_lds_b32(
        (as1_int*)(gth + i), (as3_int*)&tt[threadIdx.x], 0, 0);
#else
    float x  = __builtin_nontemporal_load(gx + i);
    float y  = __builtin_nontemporal_load(gy + i);
    float th = __builtin_nontemporal_load(gth + i);
#endif

    // Lookahead prefetch of the streaming inputs (global_prefetch_b8).
    const unsigned PF = BLK * 64u;
    if (i + PF < (unsigned)n) {
        __builtin_prefetch(gx + i + PF, 0, 1);
        __builtin_prefetch(gy + i + PF, 0, 1);
        __builtin_prefetch(gth + i + PF, 0, 1);
    }

    // -------- Stage 2: index-only RNG work overlaps the in-flight loads --------
    // Gumbel noise, JAX counter layout for shape [N,3]: flat t = 3i+j,
    // half = 3N/2; threefry pair (b, b+half), out0 for t<half else out1.
    const unsigned h3 = (3u * (unsigned)n) >> 1;
    const bool lo3 = i < ((unsigned)n >> 1);          // 3i+j < h3  <=>  i < N/2
    const unsigned base = lo3 ? 3u * i : 3u * i - h3;
    U2 r0 = tf2x32(KCAT.a, KCAT.b, base + 0u, base + 0u + h3);
    U2 r1 = tf2x32(KCAT.a, KCAT.b, base + 1u, base + 1u + h3);
    U2 r2 = tf2x32(KCAT.a, KCAT.b, base + 2u, base + 2u + h3);
    const float TINY = 1.17549435e-38f;
    float u0 = fmaxf(TINY, u01(lo3 ? r0.a : r0.b));
    float u1 = fmaxf(TINY, u01(lo3 ? r1.a : r1.b));
    float u2 = fmaxf(TINY, u01(lo3 ? r2.a : r2.b));
    float g0 = -__logf(-__logf(u0));
    float g1 = -__logf(-__logf(u1));
    float g2 = -__logf(-__logf(u2));

    // Theta resample uniform (shape [N]: half = N/2, pair (b, b+half)).
    const unsigned h = (unsigned)n >> 1;
    const bool lot = i < h;
    const unsigned bt = lot ? i : i - h;
    U2 rt = tf2x32(KTH.a, KTH.b, bt, bt + h);
    float trand = u01(lot ? rt.a : rt.b) * 6.283184f;   // TWO_PI = 2*3.141592

#if USE_ASYNC_LDS
    // Scheduling fence: force the RNG block to be materialized BEFORE the
    // asynccnt wait, so ~280 VALU ops (4x threefry + gumbel logs) execute in
    // the shadow of the in-flight async loads. The volatile asm cannot be
    // reordered against the side-effecting wait intrinsic below.
    asm volatile("" : "+v"(g0), "+v"(g1), "+v"(g2), "+v"(trand));
    // -------- Stage 3: consume staged inputs (same-wave producer/consumer) --------
    __builtin_amdgcn_s_wait_asynccnt(0);
    float x  = tx[threadIdx.x];
    float y  = ty[threadIdx.x];
    float th = tt[threadIdx.x];
#endif

    // ---- sensor directions (hardware v_sin/v_cos) ----
    const float OFF = 0.6f;
    float sl, cl, sr, cr, sc, cc;
    __sincosf(th - OFF, &sl, &cl);
    __sincosf(th + OFF, &sr, &cr);
    __sincosf(th,       &sc, &cc);

    // ---- sensor positions, clipped; truncating int cast; gather (RT policy, L2-resident) ----
    const float HI = 4095.0f;
    float sxl = fminf(fmaxf(x + cl, 0.0f), HI), syl = fminf(fmaxf(y + sl, 0.0f), HI);
    float sxr = fminf(fmaxf(x + cr, 0.0f), HI), syr = fminf(fmaxf(y + sr, 0.0f), HI);
    float sxc = fminf(fmaxf(x + cc, 0.0f), HI), syc = fminf(fmaxf(y + sc, 0.0f), HI);
    float dl = frame[((int)sxl << 12) + (int)syl];
    float dr = frame[((int)sxr << 12) + (int)syr];
    float dc = frame[((int)sxc << 12) + (int)syc];

    // ---- softmax over the 3 detections (these probs are the categorical "logits") ----
    float m  = fmaxf(dl, fmaxf(dr, dc));
    float e0 = __expf(dl - m), e1 = __expf(dr - m), e2 = __expf(dc - m);
    float inv = 1.0f / (e0 + e1 + e2);
    float p0 = e0 * inv, p1 = e1 * inv, p2 = e2 * inv;

    // ---- Gumbel-argmax (first-max tie rule, like jnp.argmax) selects direction ----
    float s0 = p0 + g0, s1 = p1 + g1, s2 = p2 + g2;
    float dx = cl, dy = sl;
    float best = s0;
    if (s1 > best) { best = s1; dx = cr; dy = sr; }
    if (s2 > best) {            dx = cc; dy = sc; }

    // ---- position update + boundary handling ----
    float xn = x + dx;   // MOVE_SPEED = 1.0
    float yn = y + dy;
    bool hx1 = xn >= 4096.0f, hx0 = xn <= 0.0f;
    bool hy1 = yn >= 4096.0f, hy0 = yn <= 0.0f;
    float xc = (hx1 || hx0) ? fminf(fmaxf(xn, 0.0f), HI) : xn;
    float yc = (hy1 || hy0) ? fminf(fmaxf(yn, 0.0f), HI) : yn;

    float hit = (float)hx1 + (float)hx0 + (float)hy1 + (float)hy0;
    float tc  = hit * trand + fabsf(hit - 1.0f) * th;   // faithful to reference formula

    // ---- NT streaming stores (protect frame's L2 residency) ----
    __builtin_nontemporal_store(xc, ox + i);
    __builtin_nontemporal_store(yc, oy + i);
    __builtin_nontemporal_store(tc, oth + i);
}

extern "C" void kernel_launch(void* const* d_in, const int* in_sizes, int n_in,
                              void* d_out, int out_size, void* d_ws, size_t ws_size,
                              hipStream_t stream) {
    (void)n_in; (void)out_size; (void)d_ws; (void)ws_size;
    const float* x     = (const float*)d_in[0];
    const float* y     = (const float*)d_in[1];
    const float* theta = (const float*)d_in[2];
    const float* frame = (const float*)d_in[3];
    float* out = (float*)d_out;
    const int n = in_sizes[0];             // N = 8388608 (multiple of BLK)
    const int blocks = (n + BLK - 1) / BLK;
    AgentUpdate_kernel<<<blocks, BLK, 0, stream>>>(
        x, y, theta, frame, out, out + n, out + 2 * n, n);
}